// NormalizedHistogram_85787676770927
// MI455X (gfx1250) — compile-verified
//
#include <hip/hip_runtime.h>

// ---- problem constants (match reference) ----
#define BATCH   32
#define IMG_H   512
#define IMG_W   512
#define CH      3
#define NB      257              // nbins + 1 bins over [0,1]
#define NBINS   256              // output bins (bin 0 dropped)
#define HW      (IMG_H * IMG_W)              // 262144 pixels per (b,c)
#define FLOATS_PER_IMG (HW * CH)             // 786432
#define VEC4_PER_IMG   (FLOATS_PER_IMG / 4)  // 196608
#define BLOCKS_PER_IMG 16
#define VEC4_PER_BLOCK (VEC4_PER_IMG / BLOCKS_PER_IMG)    // 12288 float4 / block
#define CHUNKS_PER_BLOCK (VEC4_PER_BLOCK / 3)             // 4096 12-float chunks
#define THREADS 256              // 8 wave32 waves per block
#define HIST_SZ (CH * NB)        // 771 u32 counters (3084 B LDS)
#define PF_DIST 4                // prefetch 4 strided iters ahead (~49 KB/block)

// Pass 0: zero the global counter array in workspace (re-run every call;
// harness poisons d_ws and the graph is replayed).
__global__ void nh_zero_kernel(unsigned int* __restrict__ counts, int n) {
    int i = blockIdx.x * blockDim.x + threadIdx.x;
    if (i < n) counts[i] = 0u;
}

// Pass 1: privatized LDS histograms, merged to global with atomics.
// Each thread consumes 12 consecutive floats = 4 whole RGB pixels, so the
// channel of each unrolled element is a compile-time constant (j % 3) and the
// c*NB*4 byte offset folds into the DS-instruction immediate offset.
__global__ void __launch_bounds__(THREADS)
nh_hist_kernel(const float* __restrict__ in, unsigned int* __restrict__ counts) {
    __shared__ unsigned int lhist[HIST_SZ];

    const int b   = blockIdx.x / BLOCKS_PER_IMG;
    const int blk = blockIdx.x % BLOCKS_PER_IMG;

    for (int i = threadIdx.x; i < HIST_SZ; i += THREADS) lhist[i] = 0u;
    __syncthreads();

    const float4* __restrict__ src =
        (const float4*)(in + (size_t)b * FLOATS_PER_IMG) + (size_t)blk * VEC4_PER_BLOCK;

    // bw = fp32(1/257) exactly as the reference's bin_width.
    const float bw = 1.0f / 257.0f;
    // c = RN(1/bw) — folded to a literal at compile time (double math is
    // constant-folded; no runtime f64 ops).
    const float c = (float)(1.0 / (double)(1.0f / 257.0f));

    for (int i = threadIdx.x; i < CHUNKS_PER_BLOCK; i += THREADS) {
        int pf = i + PF_DIST * THREADS;
        if (pf < CHUNKS_PER_BLOCK)
            __builtin_prefetch((const void*)(src + (size_t)pf * 3), 0, 3);

        const float4* p = src + (size_t)i * 3;     // 48 B = 4 RGB pixels
        float4 v0 = p[0];                          // global_load_b128 x3
        float4 v1 = p[1];
        float4 v2 = p[2];
        float vals[12] = {v0.x, v0.y, v0.z, v0.w,
                          v1.x, v1.y, v1.z, v1.w,
                          v2.x, v2.y, v2.z, v2.w};
        #pragma unroll
        for (int j = 0; j < 12; ++j) {
            // Correctly-rounded x/bw via Markstein (mul + 2 FMA): bit-identical
            // to IEEE RN fp32 division, matching floor(x / bin_width) exactly.
            float x  = vals[j];
            float q0 = x * c;
            float r  = fmaf(-bw, q0, x);   // exact residual (FMA)
            float q  = fmaf(r, c, q0);     // = RN(x / bw)
            int bin  = (int)q;             // trunc == floor (q >= 0)
            bin = bin > NB - 1 ? NB - 1 : bin;   // clip (q can round to 257.0)
            atomicAdd(&lhist[(j % 3) * NB + bin], 1u);  // ds_add_u32, imm offset
        }
    }
    __syncthreads();

    unsigned int* __restrict__ dst = counts + (size_t)b * HIST_SZ;
    for (int i = threadIdx.x; i < HIST_SZ; i += THREADS) {
        unsigned int v = lhist[i];
        if (v) atomicAdd(&dst[i], v);              // global_atomic_add_u32
    }
}

// Pass 2: drop bin 0, normalize per (b,c), transpose to [B, nbins, C].
// sum(hist[1..256]) == HW - count[bin0] exactly (integer counts <= 2^24, so the
// reference's fp32 summation is exact and equals this).
__global__ void __launch_bounds__(NBINS)
nh_norm_kernel(const unsigned int* __restrict__ counts, float* __restrict__ out) {
    const int b   = blockIdx.x;   // image
    const int bin = threadIdx.x;  // 0..255
    const unsigned int* __restrict__ cb = counts + (size_t)b * HIST_SZ;
    #pragma unroll
    for (int ch = 0; ch < CH; ++ch) {
        float denom = fmaxf((float)(HW - cb[ch * NB]), 1e-7f);
        float h     = (float)cb[ch * NB + bin + 1];
        out[((size_t)b * NBINS + bin) * CH + ch] = h / denom;
    }
}

extern "C" void kernel_launch(void* const* d_in, const int* in_sizes, int n_in,
                              void* d_out, int out_size, void* d_ws, size_t ws_size,
                              hipStream_t stream) {
    const float* input = (const float*)d_in[0];      // [32,512,512,3] fp32
    float* out = (float*)d_out;                      // [32,256,3] fp32
    unsigned int* counts = (unsigned int*)d_ws;      // [32][3][257] u32 (~98.7 KB)

    const int ncounts = BATCH * HIST_SZ;

    nh_zero_kernel<<<(ncounts + THREADS - 1) / THREADS, THREADS, 0, stream>>>(counts, ncounts);
    nh_hist_kernel<<<BATCH * BLOCKS_PER_IMG, THREADS, 0, stream>>>(input, counts);
    nh_norm_kernel<<<BATCH, NBINS, 0, stream>>>(counts, out);
}